// NonAttentiveTacotron_48868137894549
// MI455X (gfx1250) — compile-verified
//
#include <hip/hip_runtime.h>

// Problem constants from the reference: B=32, S=512, T=8192, C=128.
#define BB 32
#define SS 512
#define TT 8192
#define CC 128
#define FPB 64   // frames per block (each thread handles FPB/16 = 4 frames)

// ---------------------------------------------------------------------------
// Kernel 1: per-batch inclusive scan of durations -> rounded int32 cumsum.
// One block (512 threads) per batch row; Hillis-Steele scan in LDS.
// ---------------------------------------------------------------------------
__global__ __launch_bounds__(SS) void cumdur_kernel(const float* __restrict__ dur,
                                                    int* __restrict__ cum) {
    __shared__ float sc[SS];
    const int b = blockIdx.x;
    const int s = threadIdx.x;
    sc[s] = dur[b * SS + s];
    __syncthreads();
#pragma unroll
    for (int off = 1; off < SS; off <<= 1) {
        float v = sc[s];
        float add = (s >= off) ? sc[s - off] : 0.0f;
        __syncthreads();
        sc[s] = v + add;
        __syncthreads();
    }
    // jnp.round = round-half-to-even = rintf under default RNE mode.
    cum[b * SS + s] = (int)rintf(sc[s]);
}

// ---------------------------------------------------------------------------
// Kernel 2: bandwidth kernel. Block = (32 lanes, 16 rows) = 512 threads,
// covering 64 frames (each thread iterates 4 frames). grid = (T/64, B).
// Per frame, one wave writes all 128 channels as 4-channel float4s ->
// coalesced 512B global_store_b128 rows. The batch's 2KB cumdur row is
// staged to LDS once per block via the CDNA5 async-to-LDS engine.
// cos is computed as sin(x + pi/2) so sin/cos lanes don't diverge.
// ---------------------------------------------------------------------------
__global__ __launch_bounds__(SS) void posenc_kernel(const int* __restrict__ cum,
                                                    float* __restrict__ out) {
    __shared__ __attribute__((aligned(16))) int s_cum[SS];
    __shared__ float s_invf[CC / 2];

    const int b    = blockIdx.y;
    const int lane = threadIdx.x;            // 0..31  : channel group
    const int trow = threadIdx.y;            // 0..15  : frame row within tile
    const int tid  = trow * 32 + lane;       // flat id; wave = 32 consecutive

    // Stage cumdur row: 128 lanes x 16B = 2KB via the async DMA path.
    // tid<128 covers exactly waves 0..3 (EXEC all-ones in issuing waves).
    if (tid < SS / 4) {
        unsigned   lds_off = (unsigned)(size_t)&s_cum[tid * 4];
        const int* gptr    = cum + b * SS + tid * 4;
        asm volatile("global_load_async_to_lds_b128 %0, %1, off"
                     :: "v"(lds_off), "v"(gptr)
                     : "memory");
    }
    // inv_freq[k] = exp(-ln(10000)/64 * k), computed once per block.
    if (tid < CC / 2) {
        s_invf[tid] = __expf((float)tid * (-9.210340371976184f / (float)(CC / 2)));
    }
    asm volatile("s_wait_asynccnt 0x0" ::: "memory");
    __syncthreads();

    const int total = s_cum[SS - 1];
    const int cbase = lane * 4;                         // 0,4,...,124
    const int k     = cbase & (CC / 2 - 1);             // channel mod 64
    // cos half gets a +pi/2 phase: cos(x) = sin(x + pi/2). Branchless.
    const float phase = (cbase & (CC / 2)) ? 1.57079632679489662f : 0.0f;
    const float f0 = s_invf[k + 0];
    const float f1 = s_invf[k + 1];
    const float f2 = s_invf[k + 2];
    const float f3 = s_invf[k + 3];

    const int tbase = blockIdx.x * FPB;

#pragma unroll
    for (int j = 0; j < FPB / 16; ++j) {
        const int t = tbase + trow + 16 * j;

        // tok = searchsorted(cumdur, t, 'right') -> first idx with cd[idx] > t
        int lo = 0, hi = SS;
        while (lo < hi) {
            int mid = (lo + hi) >> 1;
            if (s_cum[mid] <= t) lo = mid + 1; else hi = mid;
        }
        const int   startv = (lo > 0) ? s_cum[lo - 1] : 0;
        const bool  valid  = (t < total);
        const float vmul   = valid ? 1.0f : 0.0f;
        // Keep native sin inside its valid domain for masked-out frames.
        const float posf   = valid ? (float)(t - startv) : 0.0f;

        float4 r;
        r.x = __sinf(posf * f0 + phase) * vmul;
        r.y = __sinf(posf * f1 + phase) * vmul;
        r.z = __sinf(posf * f2 + phase) * vmul;
        r.w = __sinf(posf * f3 + phase) * vmul;

        const size_t off = (((size_t)b * TT) + (size_t)t) * CC + (size_t)cbase;
        *(float4*)(out + off) = r;                      // global_store_b128
    }
}

// ---------------------------------------------------------------------------
// Launch: scan -> posenc, both on `stream`. d_ws holds cumdur (64KB int32).
// ---------------------------------------------------------------------------
extern "C" void kernel_launch(void* const* d_in, const int* in_sizes, int n_in,
                              void* d_out, int out_size, void* d_ws, size_t ws_size,
                              hipStream_t stream) {
    const float* dur = (const float*)d_in[0];   // durations [B,S] f32
    float*       out = (float*)d_out;           // [B,T,C] f32
    int*         cum = (int*)d_ws;              // [B,S] i32 scratch

    cumdur_kernel<<<BB, SS, 0, stream>>>(dur, cum);

    dim3 blk(32, 16);
    dim3 grd(TT / FPB, BB);
    posenc_kernel<<<grd, blk, 0, stream>>>(cum, out);
}